// MemoryWithCapacityManagement_16071767622140
// MI455X (gfx1250) — compile-verified
//
#include <hip/hip_runtime.h>
#include <hip/hip_bf16.h>
#include <math.h>

// ---- problem constants (match reference) ----
#define LR        0.1f
#define MAX_NORM  10.0f
#define PRUNE_T   0.01f
#define EPS_K     1e-6f

#define BATCH   8
#define SEQ     128
#define DKDIM   512
#define DVDIM   512

// ---- mapping ----
#define NWG     8                 // workgroups per batch (row-split of M)
#define ROWS    (DVDIM / NWG)     // 64 rows per WG -> 64*512*4 = 128 KB LDS slab
#define THREADS 256               // 8 waves (wave32)
#define WAVES   (THREADS / 32)

typedef float v2f __attribute__((ext_vector_type(2)));
typedef float v8f __attribute__((ext_vector_type(8)));

// Zero the cross-workgroup reduction slots (counters must start at 0).
__global__ void ws_zero_kernel(unsigned int* __restrict__ ws, int n) {
    int i = blockIdx.x * blockDim.x + threadIdx.x;
    if (i < n) ws[i] = 0u;
}

// ---- matvec tile: 16 rows x (DK/2) columns of eff(M) dotted with k_n ----
// One wave; B operand is k_n broadcast across all 16 WMMA columns so column 0
// of D accumulates the dot product. PRUNE specializes out the predicate on the
// 115/128 steps where pruning is inactive.
template<bool PRUNE>
__device__ __forceinline__ v8f matvec_tile(const float* __restrict__ Arow,
                                           const float* __restrict__ knp,
                                           int kbase, int kgrp, float tLocal) {
    v8f acc = {0.f,0.f,0.f,0.f,0.f,0.f,0.f,0.f};
    #pragma unroll 4
    for (int k4 = 0; k4 < DKDIM / 2; k4 += 4) {
        int kk = kbase + k4 + kgrp;
        float a0 = Arow[kk], a1 = Arow[kk + 1];
        v2f a, b;
        if constexpr (PRUNE) {
            a.x = (fabsf(a0) > tLocal) ? a0 : 0.0f;
            a.y = (fabsf(a1) > tLocal) ? a1 : 0.0f;
        } else {
            a.x = a0;
            a.y = a1;
        }
        b.x = knp[kk];
        b.y = knp[kk + 1];
        acc = __builtin_amdgcn_wmma_f32_16x16x4_f32(false, a, false, b,
                                                    (short)0, acc, false, false);
    }
    return acc;
}

// ---- rank-1 update over 2 rows x 64 cols; materializes lazy scale (+prune),
// returns this thread's sum-of-squares of the updated entries. ----
template<bool PRUNE>
__device__ __forceinline__ float update_rows(float4* __restrict__ m0,
                                             float4* __restrict__ m1,
                                             const float4* __restrict__ kp,
                                             float u0, float u1,
                                             float cCur, float tLocal) {
    float ssq = 0.0f;
    #pragma unroll 4
    for (int j = 0; j < 16; ++j) {
        float4 kq = kp[j];
        float4 a4 = m0[j];
        float4 b4 = m1[j];
        if constexpr (PRUNE) {
            a4.x = (fabsf(a4.x) > tLocal) ? a4.x : 0.f;
            a4.y = (fabsf(a4.y) > tLocal) ? a4.y : 0.f;
            a4.z = (fabsf(a4.z) > tLocal) ? a4.z : 0.f;
            a4.w = (fabsf(a4.w) > tLocal) ? a4.w : 0.f;
            b4.x = (fabsf(b4.x) > tLocal) ? b4.x : 0.f;
            b4.y = (fabsf(b4.y) > tLocal) ? b4.y : 0.f;
            b4.z = (fabsf(b4.z) > tLocal) ? b4.z : 0.f;
            b4.w = (fabsf(b4.w) > tLocal) ? b4.w : 0.f;
        }
        a4.x = fmaf(cCur, a4.x, -u0 * kq.x);
        a4.y = fmaf(cCur, a4.y, -u0 * kq.y);
        a4.z = fmaf(cCur, a4.z, -u0 * kq.z);
        a4.w = fmaf(cCur, a4.w, -u0 * kq.w);
        b4.x = fmaf(cCur, b4.x, -u1 * kq.x);
        b4.y = fmaf(cCur, b4.y, -u1 * kq.y);
        b4.z = fmaf(cCur, b4.z, -u1 * kq.z);
        b4.w = fmaf(cCur, b4.w, -u1 * kq.w);
        m0[j] = a4; m1[j] = b4;
        ssq = fmaf(a4.x, a4.x, ssq); ssq = fmaf(a4.y, a4.y, ssq);
        ssq = fmaf(a4.z, a4.z, ssq); ssq = fmaf(a4.w, a4.w, ssq);
        ssq = fmaf(b4.x, b4.x, ssq); ssq = fmaf(b4.y, b4.y, ssq);
        ssq = fmaf(b4.z, b4.z, ssq); ssq = fmaf(b4.w, b4.w, ssq);
    }
    return ssq;
}

__global__ __launch_bounds__(THREADS, 1)
void titans_scan_kernel(const float* __restrict__ mem_in,
                        const float* __restrict__ key,
                        const float* __restrict__ val,
                        float* __restrict__ out,
                        float* __restrict__ ws_sums,        // [BATCH*SEQ*NWG]
                        unsigned int* __restrict__ ws_cnt)  // [BATCH*SEQ]
{
    __shared__ float Mlds[ROWS * DKDIM];   // 128 KB state slab, resident all 128 steps
    __shared__ float kn  [DKDIM];          // normalized key
    __shared__ float ured[WAVES * 16];     // per-wave matvec partials (tile x khalf)
    __shared__ float ush [ROWS];           // u = (1+LR)*pred - LR*v
    __shared__ float vsh [ROWS];
    __shared__ float wred[WAVES];
    __shared__ float bc  [1];              // next scale factor c

    const int tid  = threadIdx.x;
    const int lane = tid & 31;
    const int wave = tid >> 5;
    const int wg   = blockIdx.x;          // 0..63
    const int b    = wg / NWG;
    const int slab = wg % NWG;
    const int row0 = slab * ROWS;

    // ---- load M slab: global -> LDS (float4) ----
    const float4* gsl = (const float4*)(mem_in + ((size_t)b * DVDIM + row0) * DKDIM);
    float4*       lsl = (float4*)Mlds;
    const int NV4 = ROWS * DKDIM / 4;     // 8192
    for (int i = tid; i < NV4; i += THREADS) lsl[i] = gsl[i];

    // Lazy-scale representation: effective M = cCur * prune(stored M),
    // prune keeps entries with |m| > tLocal.
    float cCur   = 1.0f;
    float tLocal = -1.0f;
    bool  pruneAct = false;

    // WMMA tiling: wave -> (16-row tile, K half)
    const int tile  = wave >> 1;                 // 0..3
    const int khalf = wave & 1;                  // 0..1
    const int kbase = khalf * (DKDIM / 2);
    const int mloc  = lane & 15;                 // A layout: lanes 0-15 = M rows
    const int kgrp  = (lane >> 4) * 2;           // A layout: K +0/+1 vs +2/+3 per lane half

    const float* Arow = Mlds + (size_t)(tile * 16 + mloc) * DKDIM;

    // update-pass addressing: thread -> 2 rows x 64 cols
    const int rp = tid >> 3;                     // 0..31
    const int cb = tid & 7;                      // 0..7
    const int r0 = rp * 2;
    float4*       mrow0 = (float4*)(Mlds + (size_t)r0 * DKDIM) + cb * 16;
    float4*       mrow1 = (float4*)(Mlds + (size_t)(r0 + 1) * DKDIM) + cb * 16;
    const float4* kblk  = (const float4*)kn + cb * 16;

    __syncthreads();

    for (int t = 0; t < SEQ; ++t) {
        // ---------- k normalization + v slab load ----------
        const float* kg = key + ((size_t)b * SEQ + t) * DKDIM;
        const float* vg = val + ((size_t)b * SEQ + t) * DVDIM + row0;
        float k0v = kg[tid];
        float k1v = kg[tid + THREADS];
        float ss  = k0v * k0v + k1v * k1v;
        for (int o = 16; o > 0; o >>= 1) ss += __shfl_down(ss, o, 32);
        if (lane == 0) wred[wave] = ss;
        if (tid < ROWS) vsh[tid] = vg[tid];
        // prefetch next step's operands
        if (t + 1 < SEQ) {
            __builtin_prefetch(kg + DKDIM + tid, 0, 0);
            if (tid < ROWS) __builtin_prefetch(vg + DVDIM + tid, 0, 0);
        }
        __syncthreads();
        float tot = 0.0f;                        // every thread, fixed order
        #pragma unroll
        for (int w = 0; w < WAVES; ++w) tot += wred[w];
        float kinv = 1.0f / (sqrtf(tot) + EPS_K);
        kn[tid]           = k0v * kinv;
        kn[tid + THREADS] = k1v * kinv;
        __syncthreads();

        // ---------- matvec via V_WMMA_F32_16X16X4_F32 ----------
        v8f acc = pruneAct ? matvec_tile<true >(Arow, kn, kbase, kgrp, tLocal)
                           : matvec_tile<false>(Arow, kn, kbase, kgrp, tLocal);
        // Extract column N==0: lane 0 holds M=r (VGPR r), lane 16 holds M=8+r.
        if ((lane & 15) == 0) {
            float* rw = ured + wave * 16 + (lane >> 4) * 8;
            #pragma unroll
            for (int r = 0; r < 8; ++r) rw[r] = acc[r];
        }
        __syncthreads();

        // ---------- combine K-halves, apply lazy scale, form u ----------
        if (tid < ROWS) {
            int tl = tid >> 4, ml = tid & 15;
            float pr = cCur * (ured[(tl * 2 + 0) * 16 + ml] + ured[(tl * 2 + 1) * 16 + ml]);
            ush[tid] = (1.0f + LR) * pr - LR * vsh[tid];   // pred + LR*(pred - v)
        }
        __syncthreads();

        // ---------- rank-1 update + materialize scale/prune + Frobenius partial ----------
        {
            float u0 = ush[r0], u1 = ush[r0 + 1];
            float ssq = pruneAct
                ? update_rows<true >(mrow0, mrow1, kblk, u0, u1, cCur, tLocal)
                : update_rows<false>(mrow0, mrow1, kblk, u0, u1, cCur, tLocal);
            for (int o = 16; o > 0; o >>= 1) ssq += __shfl_down(ssq, o, 32);
            if (lane == 0) wred[wave] = ssq;
        }
        __syncthreads();

        // ---------- cross-workgroup Frobenius reduction (deterministic slots) ----------
        if (tid == 0) {
            float part = 0.0f;
            #pragma unroll
            for (int w = 0; w < WAVES; ++w) part += wred[w];
            int slot = b * SEQ + t;
            __hip_atomic_store(&ws_sums[slot * NWG + slab], part,
                               __ATOMIC_RELEASE, __HIP_MEMORY_SCOPE_AGENT);
            __hip_atomic_fetch_add(&ws_cnt[slot], 1u,
                                   __ATOMIC_RELEASE, __HIP_MEMORY_SCOPE_AGENT);
            while (__hip_atomic_load(&ws_cnt[slot], __ATOMIC_ACQUIRE,
                                     __HIP_MEMORY_SCOPE_AGENT) < (unsigned)NWG)
                __builtin_amdgcn_s_sleep(1);
            float totq = 0.0f;
            for (int w = 0; w < NWG; ++w)       // fixed order -> deterministic
                totq += __hip_atomic_load(&ws_sums[slot * NWG + w],
                                          __ATOMIC_RELAXED, __HIP_MEMORY_SCOPE_AGENT);
            float nrm = sqrtf(totq);
            bc[0] = (nrm > MAX_NORM) ? (MAX_NORM / nrm) : 1.0f;  // c = 1/max(1, nrm/10)
        }
        __syncthreads();
        cCur     = bc[0];
        pruneAct = ((t % 10) == 0);              // prune applies to next step's reads
        tLocal   = pruneAct ? (PRUNE_T / cCur) : -1.0f;
        __syncthreads();
    }

    // ---------- write final M = cCur * prune(stored); t=127 -> prune inactive ----------
    float4* og = (float4*)(out + ((size_t)b * DVDIM + row0) * DKDIM);
    for (int i = tid; i < NV4; i += THREADS) {
        float4 m = lsl[i];
        m.x = cCur * ((fabsf(m.x) > tLocal) ? m.x : 0.f);
        m.y = cCur * ((fabsf(m.y) > tLocal) ? m.y : 0.f);
        m.z = cCur * ((fabsf(m.z) > tLocal) ? m.z : 0.f);
        m.w = cCur * ((fabsf(m.w) > tLocal) ? m.w : 0.f);
        og[i] = m;
    }
}

extern "C" void kernel_launch(void* const* d_in, const int* in_sizes, int n_in,
                              void* d_out, int out_size, void* d_ws, size_t ws_size,
                              hipStream_t stream) {
    (void)in_sizes; (void)n_in; (void)out_size; (void)ws_size;
    const float* mem = (const float*)d_in[0];
    const float* key = (const float*)d_in[1];
    const float* val = (const float*)d_in[2];
    float*       out = (float*)d_out;

    float*        sums = (float*)d_ws;                              // 8*128*8 floats
    unsigned int* cnts = (unsigned int*)d_ws + BATCH * SEQ * NWG;   // 8*128 uints

    int zwords = BATCH * SEQ * NWG + BATCH * SEQ;   // 9216 words (~36 KB)
    ws_zero_kernel<<<(zwords + 255) / 256, 256, 0, stream>>>((unsigned int*)d_ws, zwords);
    titans_scan_kernel<<<BATCH * NWG, THREADS, 0, stream>>>(mem, key, val, out, sums, cnts);
}